// AN_model_35485019800326
// MI455X (gfx1250) — compile-verified
//
#include <hip/hip_runtime.h>
#include <math.h>

typedef __attribute__((ext_vector_type(2))) float v2f;
typedef __attribute__((ext_vector_type(8))) float v8f;

#define N_ROADS 512
#define A_MAXN  8
#define BATCH   8
#define EDGE_D  8
#define OUT_D   16
#define SRC_D   64
#define OBS_D   32
#define HID_D   64

// ---------------------------------------------------------------------------
// Wave32 f32 WMMA tile: acc += A[16xK] @ B[Kx16], K multiple of 4.
// A layout (ISA 32-bit A 16x4): lanes 0-15 -> rows, VGPR0/1 = K, K+1 ; lanes
// 16-31 -> K+2, K+3.  B is the transposed analog (rows striped across lanes).
// D/C: VGPR v, lane<16 -> D[v][lane]; lane>=16 -> D[v+8][lane-16].
// ---------------------------------------------------------------------------
__device__ __forceinline__ v8f wmma_gemm_tile(const float* __restrict__ A, int lda,
                                              const float* __restrict__ Bm, int ldb,
                                              int K, v8f acc) {
    const int lane = threadIdx.x & 31;
    const int half = lane >> 4;
    const int l    = lane & 15;
    for (int k0 = 0; k0 < K; k0 += 4) {
        v2f a, b;
        a.x = A[l * lda + k0 + 2 * half];
        a.y = A[l * lda + k0 + 2 * half + 1];
        b.x = Bm[(k0 + 2 * half) * ldb + l];
        b.y = Bm[(k0 + 2 * half + 1) * ldb + l];
        acc = __builtin_amdgcn_wmma_f32_16x16x4_f32(false, a, false, b,
                                                    (short)0, acc, false, false);
    }
    return acc;
}

__device__ __forceinline__ void store_tile(float* __restrict__ D, int ldd, v8f acc) {
    const int lane = threadIdx.x & 31;
    const int half = lane >> 4;
    const int l    = lane & 15;
#pragma unroll
    for (int v = 0; v < 8; ++v)
        D[(v + 8 * half) * ldd + l] = acc[v];
}

// ---------------------------------------------------------------------------
// x = obs_all @ W_obs + b_obs      [B*N,32] x [32,16] -> [B*N,16]
// one wave per 16-row tile
// ---------------------------------------------------------------------------
__global__ void k_x(const float* __restrict__ obs_all, const float* __restrict__ W_obs,
                    const float* __restrict__ b_obs, float* __restrict__ x) {
    const int r0 = blockIdx.x * 16;
    v8f acc = {};
    acc = wmma_gemm_tile(obs_all + (size_t)r0 * OBS_D, OBS_D, W_obs, OUT_D, OBS_D, acc);
    const float bias = b_obs[threadIdx.x & 15];
#pragma unroll
    for (int v = 0; v < 8; ++v) acc[v] += bias;
    store_tile(x + (size_t)r0 * OUT_D, OUT_D, acc);
}

// ---------------------------------------------------------------------------
// y1 = x @ W_emb1[:16], y2 = x @ W_emb1[16:32]   (iter-1 weights only; iter 0 is dead)
// ---------------------------------------------------------------------------
__global__ void k_y12(const float* __restrict__ x, const float* __restrict__ W_emb,
                      float* __restrict__ y1, float* __restrict__ y2) {
    const int r0 = blockIdx.x * 16;
    const float* W1 = W_emb + 48 * OUT_D;              // iter1 base (48*16=768)
    const float* W2 = W1 + OUT_D * OUT_D;              // rows 16..31
    v8f a1 = {}, a2 = {};
    a1 = wmma_gemm_tile(x + (size_t)r0 * OUT_D, OUT_D, W1, OUT_D, OUT_D, a1);
    a2 = wmma_gemm_tile(x + (size_t)r0 * OUT_D, OUT_D, W2, OUT_D, OUT_D, a2);
    store_tile(y1 + (size_t)r0 * OUT_D, OUT_D, a1);
    store_tile(y2 + (size_t)r0 * OUT_D, OUT_D, a2);
}

// ---------------------------------------------------------------------------
// deg[i] = sum_j adj[i,j]
// ---------------------------------------------------------------------------
__global__ void k_deg(const float* __restrict__ adj, float* __restrict__ deg) {
    int i = blockIdx.x * blockDim.x + threadIdx.x;
    if (i < N_ROADS) {
        float s = 0.f;
        for (int j = 0; j < N_ROADS; ++j) s += adj[i * N_ROADS + j];
        deg[i] = s;
    }
}

// ---------------------------------------------------------------------------
// THE bandwidth kernel: Eagg[b,i,e] = sum_j adj[i,j] * edge_attrs[b,i,j,e]
// One block per (b,i): streams 16 KB contiguous with b128 loads + prefetch.
// Cross-lane xor-tree reduction (lane bit 0 = e-group parity; xor strides
// 2/4/8/16 keep parity groups intact) -> only 2 float4 LDS writes per wave.
// ---------------------------------------------------------------------------
__global__ void k_eagg(const float* __restrict__ edge, const float* __restrict__ adj,
                       float* __restrict__ eagg) {
    __shared__ float  s_adj[N_ROADS];
    __shared__ float4 s_red[16];                       // 8 waves x 2 parities
    const int blk = blockIdx.x;
    const int b = blk >> 9;
    const int i = blk & (N_ROADS - 1);
    const int tid = threadIdx.x;
    s_adj[tid]       = adj[i * N_ROADS + tid];
    s_adj[tid + 256] = adj[i * N_ROADS + tid + 256];
    __syncthreads();
    const float4* ea = (const float4*)(edge + ((size_t)(b * N_ROADS + i)) * N_ROADS * EDGE_D);
    float a0 = 0.f, a1 = 0.f, a2 = 0.f, a3 = 0.f;
#pragma unroll
    for (int s = 0; s < 4; ++s) {
        const int idx = tid + 256 * s;                 // float4 index; 2 per j
        __builtin_prefetch((const void*)(ea + idx + 256), 0, 0);
        const float4 v = ea[idx];
        const float  w = s_adj[idx >> 1];              // adj[i, j], j = idx/2
        a0 += w * v.x; a1 += w * v.y; a2 += w * v.z; a3 += w * v.w;
    }
    // in-register reduction across the 16 same-parity lanes of this wave
#pragma unroll
    for (int m = 2; m <= 16; m <<= 1) {
        a0 += __shfl_xor(a0, m, 32);
        a1 += __shfl_xor(a1, m, 32);
        a2 += __shfl_xor(a2, m, 32);
        a3 += __shfl_xor(a3, m, 32);
    }
    const int lane = tid & 31;
    const int wv   = tid >> 5;
    if (lane < 2) s_red[wv * 2 + lane] = make_float4(a0, a1, a2, a3);
    __syncthreads();
    if (tid < 8) {
        const int p = tid >> 2;                        // 0 -> e 0..3, 1 -> e 4..7
        const int c = tid & 3;
        float s = 0.f;
#pragma unroll
        for (int w = 0; w < 8; ++w) s += ((const float*)&s_red[w * 2 + p])[c];
        eagg[((size_t)(b * N_ROADS + i)) * EDGE_D + p * 4 + c] = s;
    }
}

// ---------------------------------------------------------------------------
// Fold: W_ew = W_edge1 @ W3  (8x16), bew = b_edge1 @ W3 (16)
// ---------------------------------------------------------------------------
__global__ void k_prep(const float* __restrict__ W_edge, const float* __restrict__ b_edge,
                       const float* __restrict__ W_emb,
                       float* __restrict__ wew, float* __restrict__ bew) {
    const float* We1 = W_edge + EDGE_D * OUT_D;        // iter 1
    const float* be1 = b_edge + OUT_D;
    const float* W3  = W_emb + 48 * OUT_D + 2 * OUT_D * OUT_D;  // rows 32..47 of iter1
    const int t = threadIdx.x;
    if (t < 128) {
        const int e = t >> 4, k = t & 15;
        float s = 0.f;
        for (int o = 0; o < OUT_D; ++o) s += We1[e * OUT_D + o] * W3[o * OUT_D + k];
        wew[t] = s;
    } else if (t < 144) {
        const int k = t - 128;
        float s = 0.f;
        for (int o = 0; o < OUT_D; ++o) s += be1[o] * W3[o * OUT_D + k];
        bew[k] = s;
    }
}

// ---------------------------------------------------------------------------
// adaptive = relu( x@W_res1 + b_res1 + adj@y2 + deg*(y1 + b_emb1 + bew) + Eagg@W_ew )
// adj@y2 done with WMMA (K=512 over neighbors j); epilogue fuses the rest.
// ---------------------------------------------------------------------------
__global__ void k_adaptive(const float* __restrict__ adj, const float* __restrict__ x,
                           const float* __restrict__ y1, const float* __restrict__ y2,
                           const float* __restrict__ eagg, const float* __restrict__ deg,
                           const float* __restrict__ W_res, const float* __restrict__ b_res,
                           const float* __restrict__ b_emb, const float* __restrict__ wew,
                           const float* __restrict__ bew, float* __restrict__ adaptive) {
    const int b  = blockIdx.x >> 5;
    const int i0 = (blockIdx.x & 31) * 16;
    v8f acc = {};
    acc = wmma_gemm_tile(adj + (size_t)i0 * N_ROADS, N_ROADS,
                         y2 + (size_t)b * N_ROADS * OUT_D, OUT_D, N_ROADS, acc);
    const int lane = threadIdx.x & 31;
    const int half = lane >> 4;
    const int k    = lane & 15;                        // output feature of this lane
    const float* Wr = W_res + OUT_D * OUT_D;           // iter 1
    const float br = b_res[OUT_D + k];
    const float be = b_emb[OUT_D + k] + bew[k];
    float wr[OUT_D];
#pragma unroll
    for (int o = 0; o < OUT_D; ++o) wr[o] = Wr[o * OUT_D + k];
    float we[EDGE_D];
#pragma unroll
    for (int e = 0; e < EDGE_D; ++e) we[e] = wew[e * OUT_D + k];
#pragma unroll
    for (int v = 0; v < 8; ++v) {
        const int i = i0 + v + 8 * half;
        const float* xr = x + ((size_t)(b * N_ROADS + i)) * OUT_D;
        float r = br;
#pragma unroll
        for (int o = 0; o < OUT_D; ++o) r += xr[o] * wr[o];
        const float* er = eagg + ((size_t)(b * N_ROADS + i)) * EDGE_D;
        float ec = 0.f;
#pragma unroll
        for (int e = 0; e < EDGE_D; ++e) ec += er[e] * we[e];
        const float d = deg[i];
        const float val = acc[v] + r + ec +
                          d * (y1[((size_t)(b * N_ROADS + i)) * OUT_D + k] + be);
        adaptive[((size_t)(b * N_ROADS + i)) * OUT_D + k] = fmaxf(val, 0.f);
    }
}

// ---------------------------------------------------------------------------
// Head: src proj + neighbor gather + GLU + 2-layer MLP + action head. B=8 -> tiny.
// ---------------------------------------------------------------------------
__global__ void k_head(const float* __restrict__ obs, const float* __restrict__ adaptive,
                       const int* __restrict__ ridxs, const int* __restrict__ nbr_idx,
                       const int* __restrict__ nbr_mask,
                       const float* __restrict__ W_src, const float* __restrict__ b_src,
                       const float* __restrict__ W_gate, const float* __restrict__ b_gate,
                       const float* __restrict__ W_b1, const float* __restrict__ b_b1,
                       const float* __restrict__ W_b2, const float* __restrict__ b_b2,
                       const float* __restrict__ W_act, const float* __restrict__ b_act,
                       float* __restrict__ out) {
    __shared__ float h[BATCH][2 * OUT_D * A_MAXN];     // 8 x 256
    __shared__ float g[BATCH][2 * OUT_D];              // 8 x 32
    __shared__ float h16[BATCH][OUT_D];
    __shared__ float h64a[BATCH][HID_D];
    __shared__ float h64b[BATCH][HID_D];
    const int tid = threadIdx.x;

    // src -> h[:, 0:128]
    for (int q = tid; q < BATCH * 128; q += 256) {
        const int b = q >> 7, c = q & 127;
        float s = b_src[c];
        for (int o = 0; o < SRC_D; ++o) s += obs[b * SRC_D + o] * W_src[o * 128 + c];
        h[b][c] = s;
    }
    // neighbor gather -> h[:, 128:256]
    for (int q = tid; q < BATCH * 128; q += 256) {
        const int b = q >> 7, c = q & 127;
        const int a = c >> 4, k = c & 15;
        const int rid = ridxs[b];
        float val = -1.0f;
        if (nbr_mask[rid * A_MAXN + a] != 0) {
            const int idx = nbr_idx[rid * A_MAXN + a];
            val = adaptive[((size_t)(b * N_ROADS + idx)) * OUT_D + k];
        }
        h[b][128 + c] = val;
    }
    __syncthreads();
    // gate projection [256]->[32]
    {
        const int b = tid >> 5, c = tid & 31;
        float s = b_gate[c];
        for (int j = 0; j < 256; ++j) s += h[b][j] * W_gate[j * 32 + c];
        g[b][c] = s;
    }
    __syncthreads();
    // GLU
    if (tid < BATCH * OUT_D) {
        const int b = tid >> 4, k = tid & 15;
        const float gate = g[b][OUT_D + k];
        h16[b][k] = g[b][k] * (1.0f / (1.0f + expf(-gate)));
    }
    __syncthreads();
    // MLP layer 1: [16]->[64]
    for (int q = tid; q < BATCH * HID_D; q += 256) {
        const int b = q >> 6, c = q & 63;
        float s = b_b1[c];
        for (int o = 0; o < OUT_D; ++o) s += h16[b][o] * W_b1[o * HID_D + c];
        h64a[b][c] = fmaxf(s, 0.f);
    }
    __syncthreads();
    // MLP layer 2: [64]->[64]
    for (int q = tid; q < BATCH * HID_D; q += 256) {
        const int b = q >> 6, c = q & 63;
        float s = b_b2[c];
        for (int o = 0; o < HID_D; ++o) s += h64a[b][o] * W_b2[o * HID_D + c];
        h64b[b][c] = fmaxf(s, 0.f);
    }
    __syncthreads();
    // action head: [64]->[8]
    if (tid < BATCH * A_MAXN) {
        const int b = tid >> 3, a = tid & 7;
        float s = b_act[a];
        for (int o = 0; o < HID_D; ++o) s += h64b[b][o] * W_act[o * A_MAXN + a];
        out[b * A_MAXN + a] = s;
    }
}

extern "C" void kernel_launch(void* const* d_in, const int* in_sizes, int n_in,
                              void* d_out, int out_size, void* d_ws, size_t ws_size,
                              hipStream_t stream) {
    (void)in_sizes; (void)n_in; (void)out_size; (void)ws_size;
    // setup_inputs() insertion order; params dict expanded inline in its order.
    const float* obs     = (const float*)d_in[0];
    const float* obs_all = (const float*)d_in[1];
    const float* edge    = (const float*)d_in[2];
    const float* adj     = (const float*)d_in[3];
    const float* W_src   = (const float*)d_in[4];
    const float* b_src   = (const float*)d_in[5];
    const float* W_obs   = (const float*)d_in[6];
    const float* b_obs   = (const float*)d_in[7];
    const float* W_res   = (const float*)d_in[8];
    const float* b_res   = (const float*)d_in[9];
    const float* W_edge  = (const float*)d_in[10];
    const float* b_edge  = (const float*)d_in[11];
    const float* W_emb   = (const float*)d_in[12];
    const float* b_emb   = (const float*)d_in[13];
    const float* W_gate  = (const float*)d_in[14];
    const float* b_gate  = (const float*)d_in[15];
    const float* W_b1    = (const float*)d_in[16];
    const float* b_b1    = (const float*)d_in[17];
    const float* W_b2    = (const float*)d_in[18];
    const float* b_b2    = (const float*)d_in[19];
    const float* W_act   = (const float*)d_in[20];
    const float* b_act   = (const float*)d_in[21];
    const int* ridxs     = (const int*)d_in[22];
    const int* nbr_idx   = (const int*)d_in[23];
    const int* nbr_mask  = (const int*)d_in[24];

    float* ws   = (float*)d_ws;
    float* x    = ws;                                   // 8*512*16
    float* y1   = x    + BATCH * N_ROADS * OUT_D;       // 8*512*16
    float* y2   = y1   + BATCH * N_ROADS * OUT_D;       // 8*512*16
    float* eagg = y2   + BATCH * N_ROADS * OUT_D;       // 8*512*8
    float* deg  = eagg + BATCH * N_ROADS * EDGE_D;      // 512
    float* adap = deg  + N_ROADS;                       // 8*512*16
    float* wew  = adap + BATCH * N_ROADS * OUT_D;       // 128
    float* bew  = wew  + EDGE_D * OUT_D;                // 16

    const int tiles = BATCH * N_ROADS / 16;             // 256 wave-tiles
    k_x       <<<tiles, 32, 0, stream>>>(obs_all, W_obs, b_obs, x);
    k_y12     <<<tiles, 32, 0, stream>>>(x, W_emb, y1, y2);
    k_deg     <<<2, 256, 0, stream>>>(adj, deg);
    k_eagg    <<<BATCH * N_ROADS, 256, 0, stream>>>(edge, adj, eagg);
    k_prep    <<<1, 160, 0, stream>>>(W_edge, b_edge, W_emb, wew, bew);
    k_adaptive<<<BATCH * (N_ROADS / 16), 32, 0, stream>>>(adj, x, y1, y2, eagg, deg,
                                                          W_res, b_res, b_emb, wew, bew, adap);
    k_head    <<<1, 256, 0, stream>>>(obs, adap, ridxs, nbr_idx, nbr_mask,
                                      W_src, b_src, W_gate, b_gate,
                                      W_b1, b_b1, W_b2, b_b2, W_act, b_act,
                                      (float*)d_out);
}